// SAGE_Products_5257039970572
// MI455X (gfx1250) — compile-verified
//
#include <hip/hip_runtime.h>
#include <hip/hip_bf16.h>
#include <math.h>

// ---------------------------------------------------------------------------
// GraphSAGE forward (3x SAGEConv + BN/ReLU + log_softmax) for MI455X (gfx1250)
// Memory-bound: ~3 GB of gather/scatter traffic dominates; GEMMs (5.4 GFLOP)
// run on V_WMMA_F32_16X16X4_F32 (fp32-exact matrix pipe, wave32).
// Weight tiles staged in LDS (zero-padded) so the K-loop is ds_load -> wmma
// with no EXEC-mask divergence and no 64-bit address math.
// ---------------------------------------------------------------------------

#define GN      100000
#define GE      1600000
#define NFEAT   100
#define NHID    64
#define NCLASS  47
#define BN_EPS  1e-5f

typedef __attribute__((ext_vector_type(2))) float v2f;
typedef __attribute__((ext_vector_type(8))) float v8f;

// ---------------------------------------------------------------- utilities
__global__ __launch_bounds__(256) void k_zero(float* __restrict__ p, long n) {
    long i = (long)blockIdx.x * blockDim.x + threadIdx.x;
    long stride = (long)gridDim.x * blockDim.x;
    for (; i < n; i += stride) p[i] = 0.0f;
}

// deg[dst] += 1
__global__ __launch_bounds__(256) void k_deg_count(const int* __restrict__ dst,
                                                   float* __restrict__ deg, int e) {
    int i = blockIdx.x * blockDim.x + threadIdx.x;
    int stride = gridDim.x * blockDim.x;
    for (; i < e; i += stride) atomicAdd(&deg[dst[i]], 1.0f);
}

// deg -> 1/max(deg,1)
__global__ __launch_bounds__(256) void k_deg_inv(float* __restrict__ deg, int n) {
    int i = blockIdx.x * blockDim.x + threadIdx.x;
    int stride = gridDim.x * blockDim.x;
    for (; i < n; i += stride) deg[i] = 1.0f / fmaxf(deg[i], 1.0f);
}

// agg[dst[e], f] += x[src[e], f]   (grid-stride over E*D; D compile-time so
// the index division is a mul-shift; f contiguous per wave -> coalesced)
template <int D>
__global__ __launch_bounds__(256) void k_scatter_sum(const float* __restrict__ x,
                                                     const int* __restrict__ src,
                                                     const int* __restrict__ dst,
                                                     float* __restrict__ agg,
                                                     long e) {
    long total = e * (long)D;
    long i = (long)blockIdx.x * blockDim.x + threadIdx.x;
    long stride = (long)gridDim.x * blockDim.x;
    for (; i < total; i += stride) {
        long ed = i / D;
        int  f  = (int)(i - ed * D);
        atomicAdd(&agg[(long)dst[ed] * D + f], x[(long)src[ed] * D + f]);
    }
}

// ------------------------------------------------------------- WMMA GEMM
// out tile = (agg * deg_inv) @ Wl + self @ Wr + bias
// 4 waves/block; each wave owns one 16x16 output tile (same column tile,
// different row tiles) and shares one LDS-staged, zero-padded weight tile.
// A: 16x4 f32 (2 VGPR): lane m = lane&15 holds row m; VGPR0/1 hold
//    K = koff, koff+1 with koff = (lane>>4)*2 in each 4-wide K step.
// B mirrors A with N = lane&15.  D: VGPR r -> row (r + 8*(lane>>4)).
template <int DIN, int DOUT>
__global__ __launch_bounds__(128) void k_sage_gemm_wmma(
        const float* __restrict__ Aagg,   // [n, DIN] neighbor sums
        const float* __restrict__ Aself,  // [n, DIN] self features
        const float* __restrict__ Wl,     // [DIN, DOUT]
        const float* __restrict__ Wr,     // [DIN, DOUT]
        const float* __restrict__ bias,   // [DOUT]
        const float* __restrict__ deg_inv,// [n]
        float* __restrict__ out,          // [n, DOUT]
        int n_row_tiles) {
    __shared__ float lWl[DIN * 16];
    __shared__ float lWr[DIN * 16];

    const int t    = threadIdx.x;          // 0..127
    const int lane = t & 31;
    const int wave = t >> 5;               // 0..3
    const int col0 = blockIdx.y * 16;

    // ---- stage weight tile into LDS (zero-pad cols >= DOUT) ---------------
    for (int i = t; i < DIN * 16; i += 128) {
        const int kk  = i >> 4;            // weight row
        const int nn  = i & 15;            // col within tile
        const int col = col0 + nn;
        const bool ok = (col < DOUT);
        lWl[i] = ok ? Wl[kk * DOUT + col] : 0.0f;
        lWr[i] = ok ? Wr[kk * DOUT + col] : 0.0f;
    }
    __syncthreads();

    const int tile = blockIdx.x * 4 + wave;
    if (tile >= n_row_tiles) return;       // uniform per wave

    const int row0 = tile * 16;
    const int mn   = lane & 15;            // row for A / col for B
    const int koff = (lane >> 4) << 1;     // 0 or 2
    const int rowA = row0 + mn;
    const float dinv = deg_inv[rowA];

    v8f c = {};

    // chain 1: (agg * deg_inv) @ Wl
    const float* ap = Aagg + (size_t)rowA * DIN + koff;
    #pragma unroll
    for (int k = 0; k < DIN; k += 4) {
        v2f a, b;
        a.x = ap[k]     * dinv;
        a.y = ap[k + 1] * dinv;
        b.x = lWl[(k + koff) * 16 + mn];
        b.y = lWl[(k + koff + 1) * 16 + mn];
        c = __builtin_amdgcn_wmma_f32_16x16x4_f32(false, a, false, b,
                                                  (short)0, c, false, false);
    }
    // chain 2: self @ Wr
    const float* sp = Aself + (size_t)rowA * DIN + koff;
    #pragma unroll
    for (int k = 0; k < DIN; k += 4) {
        v2f a, b;
        a.x = sp[k];
        a.y = sp[k + 1];
        b.x = lWr[(k + koff) * 16 + mn];
        b.y = lWr[(k + koff + 1) * 16 + mn];
        c = __builtin_amdgcn_wmma_f32_16x16x4_f32(false, a, false, b,
                                                  (short)0, c, false, false);
    }

    const int col = col0 + mn;
    if (col < DOUT) {
        const float bv = bias[col];
        const int rbase = row0 + ((lane >> 4) << 3);
        #pragma unroll
        for (int r = 0; r < 8; ++r)
            out[(size_t)(rbase + r) * DOUT + col] = c[r] + bv;
    }
}

// ------------------------------------------------------------- BatchNorm
// stats[0:64]=sum, [64:128]=sumsq  (NHID == 64 channels)
__global__ __launch_bounds__(256) void k_bn_stats(const float* __restrict__ h,
                                                  float* __restrict__ stats, int n) {
    __shared__ float ssum[256];
    __shared__ float ssq[256];
    const int t   = threadIdx.x;
    const int ch  = t & 63;
    const int sub = t >> 6;                 // 0..3
    float s = 0.0f, q = 0.0f;
    for (long row = (long)blockIdx.x * 4 + sub; row < n;
         row += (long)gridDim.x * 4) {
        float v = h[row * 64 + ch];
        s += v;
        q += v * v;
    }
    ssum[t] = s; ssq[t] = q;
    __syncthreads();
    if (sub == 0) {
        s = ssum[ch] + ssum[ch + 64] + ssum[ch + 128] + ssum[ch + 192];
        q = ssq[ch]  + ssq[ch + 64]  + ssq[ch + 128]  + ssq[ch + 192];
        atomicAdd(&stats[ch], s);
        atomicAdd(&stats[64 + ch], q);
    }
}

// stats[128:192]=scale (g*rstd), [192:256]=shift (beta - mu*scale)
__global__ __launch_bounds__(64) void k_bn_finalize(float* __restrict__ stats,
                                                    const float* __restrict__ g,
                                                    const float* __restrict__ beta,
                                                    int n) {
    int ch = threadIdx.x;
    float inv_n = 1.0f / (float)n;
    float mu  = stats[ch] * inv_n;
    float var = stats[64 + ch] * inv_n - mu * mu;
    float sc  = g[ch] * rsqrtf(var + BN_EPS);
    stats[128 + ch] = sc;
    stats[192 + ch] = beta[ch] - mu * sc;
}

__global__ __launch_bounds__(256) void k_bn_apply_relu(float* __restrict__ h,
                                                       const float* __restrict__ stats,
                                                       long total) {
    long i = (long)blockIdx.x * blockDim.x + threadIdx.x;
    long stride = (long)gridDim.x * blockDim.x;
    for (; i < total; i += stride) {
        int ch = (int)(i & 63);
        h[i] = fmaxf(fmaf(h[i], stats[128 + ch], stats[192 + ch]), 0.0f);
    }
}

// ------------------------------------------------------------- log_softmax
// One wave32 per node over 47 classes (2 elems/lane, shfl_xor reductions).
__global__ __launch_bounds__(256) void k_log_softmax(float* __restrict__ out, int n) {
    const int lane = threadIdx.x & 31;
    const int wib  = threadIdx.x >> 5;
    long node = (long)blockIdx.x * (blockDim.x >> 5) + wib;
    if (node >= n) return;                  // uniform per wave
    float* row = out + node * NCLASS;
    const bool hi = (lane + 32) < NCLASS;
    float a = row[lane];
    float b = hi ? row[lane + 32] : -INFINITY;
    float mx = fmaxf(a, b);
    #pragma unroll
    for (int off = 16; off; off >>= 1)
        mx = fmaxf(mx, __shfl_xor(mx, off, 32));
    float se = __expf(a - mx) + (hi ? __expf(b - mx) : 0.0f);
    #pragma unroll
    for (int off = 16; off; off >>= 1)
        se += __shfl_xor(se, off, 32);
    float lse = mx + __logf(se);
    row[lane] = a - lse;
    if (hi) row[lane + 32] = b - lse;
}

// ---------------------------------------------------------------------------
extern "C" void kernel_launch(void* const* d_in, const int* in_sizes, int n_in,
                              void* d_out, int out_size, void* d_ws, size_t ws_size,
                              hipStream_t stream) {
    const float* x    = (const float*)d_in[0];
    const int*   src  = (const int*)  d_in[1];
    const int*   dst  = (const int*)  d_in[2];
    const float* W1l  = (const float*)d_in[3];
    const float* W1r  = (const float*)d_in[4];
    const float* b1   = (const float*)d_in[5];
    const float* g1   = (const float*)d_in[6];
    const float* be1  = (const float*)d_in[7];
    const float* W2l  = (const float*)d_in[8];
    const float* W2r  = (const float*)d_in[9];
    const float* b2   = (const float*)d_in[10];
    const float* g2   = (const float*)d_in[11];
    const float* be2  = (const float*)d_in[12];
    const float* Wol  = (const float*)d_in[13];
    const float* Wor  = (const float*)d_in[14];
    const float* bo   = (const float*)d_in[15];
    float* out = (float*)d_out;

    // workspace layout (floats)
    float* w        = (float*)d_ws;
    float* deg_inv  = w;                                // N
    float* stats    = deg_inv + GN;                     // 256
    float* agg      = stats + 256;                      // N * NFEAT (reused)
    float* h1       = agg + (long)GN * NFEAT;           // N * NHID
    float* h2       = h1  + (long)GN * NHID;            // N * NHID
    (void)ws_size; (void)n_in; (void)in_sizes; (void)out_size;

    const int  ZB = 2048;     // grid for zero / elementwise
    const dim3 blk256(256);
    const int  NT = GN / 16;                 // 6250 row tiles
    const int  GB = (NT + 3) / 4;            // blocks of 4 waves

    // ---- degrees -----------------------------------------------------------
    k_zero<<<ZB, blk256, 0, stream>>>(deg_inv, (long)GN);
    k_deg_count<<<4096, blk256, 0, stream>>>(dst, deg_inv, GE);
    k_deg_inv<<<ZB, blk256, 0, stream>>>(deg_inv, GN);

    // ---- layer 1: SAGE(100 -> 64) + BN + ReLU ------------------------------
    k_zero<<<ZB, blk256, 0, stream>>>(agg, (long)GN * NFEAT);
    k_scatter_sum<NFEAT><<<8192, blk256, 0, stream>>>(x, src, dst, agg, (long)GE);
    k_sage_gemm_wmma<NFEAT, NHID><<<dim3(GB, NHID / 16), dim3(128), 0, stream>>>(
        agg, x, W1l, W1r, b1, deg_inv, h1, NT);
    k_zero<<<1, blk256, 0, stream>>>(stats, 256L);
    k_bn_stats<<<1024, blk256, 0, stream>>>(h1, stats, GN);
    k_bn_finalize<<<1, dim3(64), 0, stream>>>(stats, g1, be1, GN);
    k_bn_apply_relu<<<ZB, blk256, 0, stream>>>(h1, stats, (long)GN * NHID);

    // ---- layer 2: SAGE(64 -> 64) + BN + ReLU -------------------------------
    k_zero<<<ZB, blk256, 0, stream>>>(agg, (long)GN * NHID);
    k_scatter_sum<NHID><<<8192, blk256, 0, stream>>>(h1, src, dst, agg, (long)GE);
    k_sage_gemm_wmma<NHID, NHID><<<dim3(GB, NHID / 16), dim3(128), 0, stream>>>(
        agg, h1, W2l, W2r, b2, deg_inv, h2, NT);
    k_zero<<<1, blk256, 0, stream>>>(stats, 256L);
    k_bn_stats<<<1024, blk256, 0, stream>>>(h2, stats, GN);
    k_bn_finalize<<<1, dim3(64), 0, stream>>>(stats, g2, be2, GN);
    k_bn_apply_relu<<<ZB, blk256, 0, stream>>>(h2, stats, (long)GN * NHID);

    // ---- output: SAGE(64 -> 47) + log_softmax ------------------------------
    k_zero<<<ZB, blk256, 0, stream>>>(agg, (long)GN * NHID);
    k_scatter_sum<NHID><<<8192, blk256, 0, stream>>>(h2, src, dst, agg, (long)GE);
    k_sage_gemm_wmma<NHID, NCLASS><<<dim3(GB, 3), dim3(128), 0, stream>>>(
        agg, h2, Wol, Wor, bo, deg_inv, out, NT);
    k_log_softmax<<<(GN + 7) / 8, blk256, 0, stream>>>(out, GN);
}